// HybridImageFittingModel_9509057593799
// MI455X (gfx1250) — compile-verified
//
#include <hip/hip_runtime.h>
#include <math.h>

typedef __attribute__((ext_vector_type(2))) float v2f;
typedef __attribute__((ext_vector_type(8))) float v8f;

#define TPB 256
#define NB_STATS 1024
#define NB_FUSED 2048

// ---------------------------------------------------------------------------
// Pass 1: raw moments of x  (Sx0, Sx1, Sx0^2, Sx1^2, Sx0x1) per block.
// Deterministic tree reduction: shfl_xor (wave32) -> LDS -> one partial/block.
// ---------------------------------------------------------------------------
__global__ void stats_kernel(const float2* __restrict__ x,
                             float* __restrict__ part, int n) {
  float s0 = 0.f, s1 = 0.f, s00 = 0.f, s11 = 0.f, s01 = 0.f;
  int idx = blockIdx.x * blockDim.x + threadIdx.x;
  int stride = gridDim.x * blockDim.x;
  for (int i = idx; i < n; i += stride) {
    float2 v = x[i];
    s0 += v.x;
    s1 += v.y;
    s00 = fmaf(v.x, v.x, s00);
    s11 = fmaf(v.y, v.y, s11);
    s01 = fmaf(v.x, v.y, s01);
  }
#pragma unroll
  for (int off = 16; off > 0; off >>= 1) {
    s0 += __shfl_xor(s0, off);
    s1 += __shfl_xor(s1, off);
    s00 += __shfl_xor(s00, off);
    s11 += __shfl_xor(s11, off);
    s01 += __shfl_xor(s01, off);
  }
  __shared__ float sm[8][5];
  int w = threadIdx.x >> 5;
  if ((threadIdx.x & 31) == 0) {
    sm[w][0] = s0; sm[w][1] = s1; sm[w][2] = s00; sm[w][3] = s11; sm[w][4] = s01;
  }
  __syncthreads();
  if (threadIdx.x < 5) {
    float acc = 0.f;
#pragma unroll
    for (int k = 0; k < 8; ++k) acc += sm[k][threadIdx.x];
    part[blockIdx.x * 5 + threadIdx.x] = acc;
  }
}

// ---------------------------------------------------------------------------
// Pass 2 (1 block): reduce partials, fold BN+linear into affine P,Q and
// precompute the two Rot-gate complex 2x2 matrices. Writes 24-float param
// block: [P00 P01 P10 P11 Q0 Q1 W2 b2 | U1(re,im x4) | U2(re,im x4)]
// ---------------------------------------------------------------------------
__global__ void finalize_kernel(const float* __restrict__ part, int nblocks,
                                const float* __restrict__ W1,
                                const float* __restrict__ b1,
                                const float* __restrict__ gamma,
                                const float* __restrict__ beta,
                                const float* __restrict__ qp,
                                const float* __restrict__ W2,
                                const float* __restrict__ b2,
                                float invn, float* __restrict__ pp) {
  float acc[5] = {0.f, 0.f, 0.f, 0.f, 0.f};
  for (int i = threadIdx.x; i < nblocks; i += blockDim.x) {
#pragma unroll
    for (int j = 0; j < 5; ++j) acc[j] += part[i * 5 + j];
  }
#pragma unroll
  for (int off = 16; off > 0; off >>= 1) {
#pragma unroll
    for (int j = 0; j < 5; ++j) acc[j] += __shfl_xor(acc[j], off);
  }
  __shared__ float sm[8][5];
  int w = threadIdx.x >> 5;
  if ((threadIdx.x & 31) == 0) {
#pragma unroll
    for (int j = 0; j < 5; ++j) sm[w][j] = acc[j];
  }
  __syncthreads();
  if (threadIdx.x == 0) {
    float tot[5] = {0.f, 0.f, 0.f, 0.f, 0.f};
#pragma unroll
    for (int k = 0; k < 8; ++k)
#pragma unroll
      for (int j = 0; j < 5; ++j) tot[j] += sm[k][j];

    float mx0 = tot[0] * invn, mx1 = tot[1] * invn;
    float vx0 = tot[2] * invn - mx0 * mx0;
    float vx1 = tot[3] * invn - mx1 * mx1;
    float cxy = tot[4] * invn - mx0 * mx1;

    float w00 = W1[0], w01 = W1[1], w10 = W1[2], w11 = W1[3];
    float mu0 = fmaf(w00, mx0, fmaf(w01, mx1, b1[0]));
    float mu1 = fmaf(w10, mx0, fmaf(w11, mx1, b1[1]));
    float var0 = w00 * w00 * vx0 + w01 * w01 * vx1 + 2.f * w00 * w01 * cxy;
    float var1 = w10 * w10 * vx0 + w11 * w11 * vx1 + 2.f * w10 * w11 * cxy;
    float r0 = 1.0f / sqrtf(var0 + 1e-5f);
    float r1 = 1.0f / sqrtf(var1 + 1e-5f);
    float A0 = gamma[0] * r0, A1 = gamma[1] * r1;
    float C0 = beta[0] - mu0 * A0, C1 = beta[1] - mu1 * A1;

    pp[0] = A0 * w00; pp[1] = A0 * w01;
    pp[2] = A1 * w10; pp[3] = A1 * w11;
    pp[4] = fmaf(A0, b1[0], C0);
    pp[5] = fmaf(A1, b1[1], C1);
    pp[6] = W2[0];
    pp[7] = b2[0];

    // Rot(phi,theta,omega) = [[e^{-i(p)}c, -e^{i(m)}s],[e^{-i(m)}s, e^{i(p)}c]]
    // with p=(phi+omega)/2, m=(phi-omega)/2, c=cos(theta/2), s=sin(theta/2)
#pragma unroll
    for (int g = 0; g < 2; ++g) {
      const float* q = qp + 3 * g;
      float* dst = pp + 8 + 8 * g;
      float c = cosf(0.5f * q[1]), s = sinf(0.5f * q[1]);
      float p = 0.5f * (q[0] + q[2]), m = 0.5f * (q[0] - q[2]);
      float sp, cp, smm, cm;
      sincosf(p, &sp, &cp);
      sincosf(m, &smm, &cm);
      dst[0] = cp * c;  dst[1] = -sp * c;   // U00
      dst[2] = -cm * s; dst[3] = -smm * s;  // U01
      dst[4] = cm * s;  dst[5] = -smm * s;  // U10
      dst[6] = cp * c;  dst[7] = sp * c;    // U11
    }
  }
}

// ---------------------------------------------------------------------------
// 2-qubit circuit on angles (a,b). State stays real through the RY/CNOT
// prefix; Rot gates U1 (wire0) and U2 (wire1) are precomputed complex 2x2.
// Final CNOT01 only swaps row-1 probabilities -> fold into the sign.
// ---------------------------------------------------------------------------
__device__ __forceinline__ float qcircuit(float a, float b,
                                          const float u1[8], const float u2[8]) {
  float sa, ca, sb, cb;
  __sincosf(0.5f * a, &sa, &ca);
  __sincosf(0.5f * b, &sb, &cb);
  // RY(a)@w0, RY(b)@w1, CNOT01:
  float r00 = ca * cb, r01 = ca * sb, r10 = sa * sb, r11 = sa * cb;
  // RY(a+b)@w1 via angle addition:
  float c2 = ca * cb - sa * sb;
  float s2 = sa * cb + ca * sb;
  float t00 = fmaf(c2, r00, -s2 * r01);
  float t01 = fmaf(s2, r00, c2 * r01);
  float t10 = fmaf(c2, r10, -s2 * r11);
  float t11 = fmaf(s2, r10, c2 * r11);
  // CNOT10: column 1 rows swap
  float g00 = t00, g01 = t11, g10 = t10, g11 = t01;
  // Rot1 on wire0 (complex gate x real state): v[i][k]=U1[i][0]g[0][k]+U1[i][1]g[1][k]
  float v00r = fmaf(u1[0], g00, u1[2] * g10), v00i = fmaf(u1[1], g00, u1[3] * g10);
  float v01r = fmaf(u1[0], g01, u1[2] * g11), v01i = fmaf(u1[1], g01, u1[3] * g11);
  float v10r = fmaf(u1[4], g00, u1[6] * g10), v10i = fmaf(u1[5], g00, u1[7] * g10);
  float v11r = fmaf(u1[4], g01, u1[6] * g11), v11i = fmaf(u1[5], g01, u1[7] * g11);
  // Rot2 on wire1 (complex x complex): w[k][i]=U2[i][0]v[k][0]+U2[i][1]v[k][1]
  float w00r = u2[0] * v00r - u2[1] * v00i + u2[2] * v01r - u2[3] * v01i;
  float w00i = u2[0] * v00i + u2[1] * v00r + u2[2] * v01i + u2[3] * v01r;
  float w01r = u2[4] * v00r - u2[5] * v00i + u2[6] * v01r - u2[7] * v01i;
  float w01i = u2[4] * v00i + u2[5] * v00r + u2[6] * v01i + u2[7] * v01r;
  float w10r = u2[0] * v10r - u2[1] * v10i + u2[2] * v11r - u2[3] * v11i;
  float w10i = u2[0] * v10i + u2[1] * v10r + u2[2] * v11i + u2[3] * v11r;
  float w11r = u2[4] * v10r - u2[5] * v10i + u2[6] * v11r - u2[7] * v11i;
  float w11i = u2[4] * v10i + u2[5] * v10r + u2[6] * v11i + u2[7] * v11r;
  // sum_i (p[i][0]-p[i][1]) after final CNOT01 (row1 swapped):
  return (w00r * w00r + w00i * w00i) - (w01r * w01r + w01i * w01i)
       + (w11r * w11r + w11i * w11i) - (w10r * w10r + w10i * w10i);
}

// ---------------------------------------------------------------------------
// Pass 3: fused affine (via V_WMMA_F32_16X16X4_F32) + circuit + output scale.
// A = P (rows 0..1 of 16x4), B = x^T for 16 rows (lanes 0..15, K=2..3 zero),
// C = Q broadcast (accumulator does the bias add). Two WMMAs + shfl_xor(16)
// give every lane of the wave its (h0,h1). Main loop keeps EXEC all-1s.
// ---------------------------------------------------------------------------
__global__ void fused_circuit_kernel(const float2* __restrict__ x,
                                     const float* __restrict__ pp,
                                     float* __restrict__ out, int n) {
  float p00 = pp[0], p01 = pp[1], p10 = pp[2], p11 = pp[3];
  float q0 = pp[4], q1 = pp[5], w2 = pp[6], bias = pp[7];
  float u1[8], u2[8];
#pragma unroll
  for (int j = 0; j < 8; ++j) { u1[j] = pp[8 + j]; u2[j] = pp[16 + j]; }

  int lane = threadIdx.x & 31;
  bool lo = lane < 16;
  // A operand (16x4): lane M holds K0 in .x, K1 in .y; lanes>=16 hold K2/K3 (=0)
  v2f aP;
  aP.x = (lane == 0) ? p00 : ((lane == 1) ? p10 : 0.f);
  aP.y = (lane == 0) ? p01 : ((lane == 1) ? p11 : 0.f);
  // C operand: row M=0 -> Q0, M=1 -> Q1 broadcast across N lanes
  v8f cQ = {};
  cQ[0] = q0;
  cQ[1] = q1;

  int idx = blockIdx.x * blockDim.x + threadIdx.x;
  int stride = gridDim.x * blockDim.x;
  int nfull = (n / stride) * stride;  // stride is a multiple of 32

  for (int i = idx; i < nfull; i += stride) {
    float2 xv = x[i];
    float sx0 = __shfl_xor(xv.x, 16);
    float sx1 = __shfl_xor(xv.y, 16);
    // B operand (4x16): VGPR0 lanes0-15 = K0 row, VGPR1 lanes0-15 = K1 row.
    v2f bA, bB;
    bA.x = lo ? xv.x : 0.f;  bA.y = lo ? xv.y : 0.f;   // rows base..base+15
    bB.x = lo ? sx0 : 0.f;   bB.y = lo ? sx1 : 0.f;    // rows base+16..base+31
    v8f d1 = __builtin_amdgcn_wmma_f32_16x16x4_f32(false, aP, false, bA,
                                                   (short)0, cQ, false, false);
    v8f d2 = __builtin_amdgcn_wmma_f32_16x16x4_f32(false, aP, false, bB,
                                                   (short)0, cQ, false, false);
    float e0 = __shfl_xor(d2[0], 16);
    float e1 = __shfl_xor(d2[1], 16);
    float h0 = lo ? d1[0] : e0;   // D: VGPR0 lane m = row m result for n=0
    float h1 = lo ? d1[1] : e1;   //    VGPR1 lane m = row m result for n=1
    float qv = qcircuit(h0, h1, u1, u2);
    out[i] = fmaf(qv, w2, bias);
  }
  // scalar tail (empty when stride | n) — keeps WMMA loop EXEC-full
  for (int i = nfull + idx; i < n; i += stride) {
    float2 xv = x[i];
    float h0 = fmaf(p00, xv.x, fmaf(p01, xv.y, q0));
    float h1 = fmaf(p10, xv.x, fmaf(p11, xv.y, q1));
    float qv = qcircuit(h0, h1, u1, u2);
    out[i] = fmaf(qv, w2, bias);
  }
}

// ---------------------------------------------------------------------------
extern "C" void kernel_launch(void* const* d_in, const int* in_sizes, int n_in,
                              void* d_out, int out_size, void* d_ws, size_t ws_size,
                              hipStream_t stream) {
  const float2* x = (const float2*)d_in[0];
  const float* W1 = (const float*)d_in[1];
  const float* b1 = (const float*)d_in[2];
  const float* gamma = (const float*)d_in[3];
  const float* beta = (const float*)d_in[4];
  const float* qp = (const float*)d_in[5];
  const float* W2 = (const float*)d_in[6];
  const float* b2 = (const float*)d_in[7];
  float* out = (float*)d_out;

  int n = out_size;  // B rows
  float* ws = (float*)d_ws;
  float* part = ws;           // NB_STATS * 5 floats
  float* pp = ws + 8192;      // 24-float param block

  stats_kernel<<<NB_STATS, TPB, 0, stream>>>(x, part, n);
  finalize_kernel<<<1, TPB, 0, stream>>>(part, NB_STATS, W1, b1, gamma, beta,
                                         qp, W2, b2, 1.0f / (float)n, pp);
  fused_circuit_kernel<<<NB_FUSED, TPB, 0, stream>>>(x, pp, out, n);
}